// AncSimilarityLoss_81003083203603
// MI455X (gfx1250) — compile-verified
//
#include <hip/hip_runtime.h>
#include <math.h>

#define Q   2048
#define DD  512
#define KC  128
#define BB  256
#define NN  (Q + BB)        // 2304 rows of "sim"
#define NK  (NN * KC)       // 294912 elements per slice
#define NBLK (NK / 256)     // 1152 blocks for elementwise passes
#define MIN_EPS 1e-5f
#define CDV 18.0f           // cd == N/K == 18 for every k (proven constant)

typedef __attribute__((ext_vector_type(2))) float v2f;
typedef __attribute__((ext_vector_type(8))) float v8f;

// ---------------- block reduction helpers (power-of-2 blockDim) ----------------
__device__ __forceinline__ float blockReduceSum(float v, float* sh) {
  int t = threadIdx.x;
  sh[t] = v; __syncthreads();
  for (int s = blockDim.x >> 1; s > 0; s >>= 1) {
    if (t < s) sh[t] += sh[t + s];
    __syncthreads();
  }
  float r = sh[0]; __syncthreads();
  return r;
}
__device__ __forceinline__ float blockReduceMax(float v, float* sh) {
  int t = threadIdx.x;
  sh[t] = v; __syncthreads();
  for (int s = blockDim.x >> 1; s > 0; s >>= 1) {
    if (t < s) sh[t] = fmaxf(sh[t], sh[t + s]);
    __syncthreads();
  }
  float r = sh[0]; __syncthreads();
  return r;
}
__device__ __forceinline__ float blockReduceMin(float v, float* sh) {
  int t = threadIdx.x;
  sh[t] = v; __syncthreads();
  for (int s = blockDim.x >> 1; s > 0; s >>= 1) {
    if (t < s) sh[t] = fminf(sh[t], sh[t + s]);
    __syncthreads();
  }
  float r = sh[0]; __syncthreads();
  return r;
}

// ---------------- 1) per-row inverse norms of w (and reset inf-max slot) ------
__global__ void k_invnorm(const float* __restrict__ w, float* __restrict__ invn,
                          unsigned int* __restrict__ Mslot) {
  int k = threadIdx.x;                       // 128 threads, one per cluster row
  float s = 0.0f;
  const float* row = w + (size_t)k * DD;
  for (int d = 0; d < DD; ++d) { float v = row[d]; s += v * v; }
  invn[k] = 1.0f / sqrtf(s);
  if (k == 0) *Mslot = 0u;
}

// ---------------- 2) sim[0:Q] = (q @ w^T) * invn  via V_WMMA_F32_16X16X4_F32 --
// One wave computes one 16x16 tile; K=4 fp32 WMMA steps over D=512.
// A 16x4 layout : lane<16 -> (A[m][0],A[m][1]); lane>=16 -> (A[m][2],A[m][3]), m=lane&15
// B 4x16 layout : lane<16 -> (B[0][n],B[1][n]); lane>=16 -> (B[2][n],B[3][n]), n=lane&15
// C/D 16x16     : vgpr r, lanes0-15 -> (M=r, N=lane); lanes16-31 -> (M=r+8, N=lane-16)
__global__ void k_gemm(const float* __restrict__ q, const float* __restrict__ w,
                       const float* __restrict__ invn, float* __restrict__ sim) {
  int lane    = threadIdx.x;                                  // 0..31 (wave32)
  int tileRow = blockIdx.y * blockDim.y + threadIdx.y;        // 0..127
  int tileCol = blockIdx.x;                                   // 0..7
  int m0 = tileRow * 16, c0 = tileCol * 16;
  int mr = lane & 15;                 // M index for A, N index for B
  int kb = (lane >> 4) << 1;          // 0 or 2 : K sub-offset within the 4-step

  const float* qa = q + (size_t)(m0 + mr) * DD + kb;   // A = q tile (row-major over D)
  const float* wb = w + (size_t)(c0 + mr) * DD + kb;   // B = w^T tile -> read w rows

  v8f acc = {};
  for (int d = 0; d < DD; d += 4) {
    v2f av, bv;
    av.x = qa[d]; av.y = qa[d + 1];
    bv.x = wb[d]; bv.y = wb[d + 1];
    acc = __builtin_amdgcn_wmma_f32_16x16x4_f32(
        /*neg_a=*/false, av, /*neg_b=*/false, bv,
        /*c_mod=*/(short)0, acc, /*reuse_a=*/false, /*reuse_b=*/false);
  }
  float scale = invn[c0 + mr];        // column scaling == row-normalized w
  int rowAdd = (lane >> 4) * 8;
#pragma unroll
  for (int r = 0; r < 8; ++r) {
    sim[(size_t)(m0 + r + rowAdd) * KC + (c0 + mr)] = acc[r] * scale;
  }
}

// ---------------- 3) lo/hi = exp(5s), exp(20s); track max over finite ---------
__global__ void k_exp_max(const float* __restrict__ sim, const float* __restrict__ a,
                          float* __restrict__ lo, float* __restrict__ hi,
                          unsigned int* __restrict__ Mslot) {
  __shared__ float sh[256];
  int idx = blockIdx.x * 256 + threadIdx.x;            // < NK
  float s = (idx < Q * KC) ? sim[idx] : a[idx - Q * KC];
  float l = expf(5.0f * s);                            // mmf=0.25 -> 0.25/0.05
  float h = expf(20.0f * s);                           // mmf=1.0  -> 1/0.05
  lo[idx] = l; hi[idx] = h;
  float mx = 0.0f;                                     // jnp.max(where(inf,0,x))
  if (!isinf(l)) mx = l;
  if (!isinf(h)) mx = fmaxf(mx, h);
  float bm = blockReduceMax(mx, sh);
  if (threadIdx.x == 0) atomicMax(Mslot, __float_as_uint(bm)); // vals>=0: uint order ok
}

// ---------------- 4) replace inf with finite max; per-block partial sums ------
__global__ void k_fix_psum(float* __restrict__ lo, float* __restrict__ hi,
                           const unsigned int* __restrict__ Mslot,
                           float* __restrict__ P) {
  __shared__ float sh[256];
  float gm = __uint_as_float(*Mslot);
  int idx = blockIdx.x * 256 + threadIdx.x;
  float l = lo[idx]; if (isinf(l)) { l = gm; lo[idx] = l; }
  float h = hi[idx]; if (isinf(h)) { h = gm; hi[idx] = h; }
  float bs = blockReduceSum(l + h, sh);
  if (threadIdx.x == 0) P[blockIdx.x] = bs;
}

// ---------------- 5) G = 64 * sum(partials)  (deterministic 2-stage sum) ------
__global__ void k_sum_final(const float* __restrict__ P, float* __restrict__ G) {
  __shared__ float sh[256];
  float v = 0.0f;
  for (int i = threadIdx.x; i < NBLK; i += 256) v += P[i];
  float tot = blockReduceSum(v, sh);
  if (threadIdx.x == 0) G[0] = 64.0f * tot;            // 64 copies of each slice
}

// ---------------- 6) A /= sum(A); column sums (axis=1) ------------------------
__global__ void k_div_colsum(float* __restrict__ lo, float* __restrict__ hi,
                             const float* __restrict__ G, float* __restrict__ CS) {
  __shared__ float sh[256];
  int bid = blockIdx.x;                  // 0..255 : slice*128 + k
  float* ptr = (bid & 128) ? hi : lo;
  int k = bid & (KC - 1);
  float inv = 1.0f / G[0];
  float s = 0.0f;
  for (int n = threadIdx.x; n < NN; n += 256) {
    float x = ptr[(size_t)n * KC + k] * inv;
    ptr[(size_t)n * KC + k] = x;
    s += x;
  }
  float tot = blockReduceSum(s, sh);
  if (threadIdx.x == 0) CS[bid] = tot;
}

// ---------------- column sums only (for iterations 2,3) -----------------------
__global__ void k_colsum(const float* __restrict__ lo, const float* __restrict__ hi,
                         float* __restrict__ CS) {
  __shared__ float sh[256];
  int bid = blockIdx.x;
  const float* ptr = (bid & 128) ? hi : lo;
  int k = bid & (KC - 1);
  float s = 0.0f;
  for (int n = threadIdx.x; n < NN; n += 256) s += ptr[(size_t)n * KC + k];
  float tot = blockReduceSum(s, sh);
  if (threadIdx.x == 0) CS[bid] = tot;
}

// ---------------- 7a+7b) A *= cd/(colsum+eps); A /= (64*(lo+hi)+eps) ----------
__global__ void k_colscale_fdiv(float* __restrict__ lo, float* __restrict__ hi,
                                const float* __restrict__ CS) {
  int idx = blockIdx.x * 256 + threadIdx.x;
  int k = idx & (KC - 1);
  float l = lo[idx] * (CDV / (CS[k] + MIN_EPS));
  float h = hi[idx] * (CDV / (CS[KC + k] + MIN_EPS));
  float t = 64.0f * (l + h);                 // axis=0 sum: 64 copies each group
  lo[idx] = l / (t + MIN_EPS);
  hi[idx] = h / (t + MIN_EPS);
}

// ---------------- 7c) row normalize (axis=2) with eps -------------------------
__global__ void k_rowdiv(float* __restrict__ lo, float* __restrict__ hi) {
  __shared__ float sh[128];
  int n = blockIdx.x, t = threadIdx.x;       // t == k
  size_t base = (size_t)n * KC;
  float l = lo[base + t];
  float rs = blockReduceSum(l, sh);
  lo[base + t] = l / (rs + MIN_EPS);
  float h = hi[base + t];
  rs = blockReduceSum(h, sh);
  hi[base + t] = h / (rs + MIN_EPS);
}

// ---------------- 8) codes = rowwise minmax(hi); cost terms -------------------
// Final A/rowsum and the x64 factor cancel inside the row min-max normalization.
__global__ void k_cost(const float* __restrict__ hi, const float* __restrict__ a,
                       float* __restrict__ PC) {
  __shared__ float sh[128];
  int r = blockIdx.x, t = threadIdx.x;       // r: assignment row, t == k
  float h = hi[(size_t)(Q + r) * KC + t];
  float ma = blockReduceMax(h, sh);
  float mi = blockReduceMin(h, sh);
  float code = (h - mi) / (ma - mi);
  float av = a[(size_t)r * KC + t];
  float x = expf(av * 10.0f);                // similarity / 0.1
  float sp = (x > 20.0f) ? x : log1pf(expf(x));   // stable softplus
  float tot = blockReduceSum(sp - code * x, sh);
  if (t == 0) PC[r] = tot;
}

__global__ void k_cost_final(const float* __restrict__ PC, float* __restrict__ Lbuf,
                             int p) {
  __shared__ float sh[256];
  float tot = blockReduceSum(PC[threadIdx.x], sh);
  if (threadIdx.x == 0) Lbuf[p] = tot / (float)(BB * KC);
}

__global__ void k_final(const float* __restrict__ Lbuf, float* __restrict__ out) {
  if (threadIdx.x == 0) {
    float s = 0.0f;
    for (int i = 0; i < 12; ++i) s += Lbuf[i];
    out[0] = s / 12.0f;                      // BETA = 1.0
  }
}

// ------------------------------------------------------------------------------
extern "C" void kernel_launch(void* const* d_in, const int* in_sizes, int n_in,
                              void* d_out, int out_size, void* d_ws, size_t ws_size,
                              hipStream_t stream) {
  (void)in_sizes; (void)n_in; (void)out_size; (void)ws_size;
  static const int qi[12] = {0, 1, 2, 3, 3, 3, 4, 4, 4, 5, 5, 5};
  static const int wi[12] = {6, 7, 8, 9, 10, 11, 9, 10, 11, 9, 10, 11};
  static const int ai[12] = {12, 13, 14, 15, 16, 17, 18, 19, 20, 21, 22, 23};

  float* W   = (float*)d_ws;
  float* sim = W;                        // Q*KC         = 262144 floats
  float* lo  = sim + (size_t)Q * KC;     // NK           = 294912
  float* hi  = lo + NK;                  // NK
  float* invn = hi + NK;                 // KC
  unsigned int* Mslot = (unsigned int*)(invn + KC);   // 1
  float* G   = (float*)(Mslot + 1);      // 1
  float* CS  = G + 1;                    // 256
  float* P   = CS + 256;                 // NBLK = 1152
  float* PC  = P + NBLK;                 // BB   = 256
  float* Lbuf = PC + BB;                 // 12

  for (int p = 0; p < 12; ++p) {
    const float* q = (const float*)d_in[qi[p]];
    const float* w = (const float*)d_in[wi[p]];
    const float* a = (const float*)d_in[ai[p]];

    k_invnorm<<<1, 128, 0, stream>>>(w, invn, Mslot);
    k_gemm<<<dim3(8, 32), dim3(32, 4), 0, stream>>>(q, w, invn, sim);
    k_exp_max<<<NBLK, 256, 0, stream>>>(sim, a, lo, hi, Mslot);
    k_fix_psum<<<NBLK, 256, 0, stream>>>(lo, hi, Mslot, P);
    k_sum_final<<<1, 256, 0, stream>>>(P, G);
    k_div_colsum<<<256, 256, 0, stream>>>(lo, hi, G, CS);
    for (int it = 0; it < 3; ++it) {
      if (it > 0) k_colsum<<<256, 256, 0, stream>>>(lo, hi, CS);
      k_colscale_fdiv<<<NBLK, 256, 0, stream>>>(lo, hi, CS);
      k_rowdiv<<<NN, 128, 0, stream>>>(lo, hi);
    }
    k_cost<<<BB, 128, 0, stream>>>(hi, a, PC);
    k_cost_final<<<1, 256, 0, stream>>>(PC, Lbuf, p);
  }
  k_final<<<1, 32, 0, stream>>>(Lbuf, (float*)d_out);
}